// GCNEncoder_27212912787783
// MI455X (gfx1250) — compile-verified
//
#include <hip/hip_runtime.h>

typedef __attribute__((ext_vector_type(2))) float v2f;
typedef __attribute__((ext_vector_type(8))) float v8f;

#define NC 64  // all hidden/output widths are 64

// ---------------------------------------------------------------------------
// Dense GEMM: H[M x 64] = (RELU ? relu(A) : A)[M x K] @ W[K x 64]
// One wave32 per 16x64 row-block: A loaded once per k-step, reused by 4
// V_WMMA_F32_16X16X4_F32 (one per 16-col tile).
// A-matrix VGPR layout (ISA 7.12.2, 32-bit A 16x4): lanes 0-15 -> K=0,1;
// lanes 16-31 -> K=2,3; M = lane&15 in both halves.
// B-matrix 4x16: VGPR0 = rows K={0 | 2}, VGPR1 = rows K={1 | 3}, N = lane&15.
// C/D 16x16: VGPR r -> M = r (+8 for lanes 16-31), N = lane&15.
// ---------------------------------------------------------------------------
template <int RELU>
__global__ __launch_bounds__(32) void gcn_gemm_wmma_f32(
    const float* __restrict__ A, const float* __restrict__ W,
    float* __restrict__ H, int M, int K)
{
  const int lane = threadIdx.x;
  const int row0 = blockIdx.x * 16;
  const int half = lane >> 4;      // 0: K-pair {0,1}, 1: K-pair {2,3}
  const int l15  = lane & 15;
  const int koff = half * 2;

  v8f c0 = {}, c1 = {}, c2 = {}, c3 = {};
  const float* __restrict__ arow  = A + (size_t)(row0 + l15) * (size_t)K;
  const float* __restrict__ wbase = W + l15;

  for (int k = 0; k < K; k += 4) {
    float a0 = arow[k + koff];
    float a1 = arow[k + koff + 1];
    if (RELU) { a0 = fmaxf(a0, 0.0f); a1 = fmaxf(a1, 0.0f); }
    v2f a = { a0, a1 };

    const float* __restrict__ w0 = wbase + (size_t)(k + koff) * NC;
    const float* __restrict__ w1 = wbase + (size_t)(k + koff + 1) * NC;
    v2f b0 = { w0[0],  w1[0]  };
    v2f b1 = { w0[16], w1[16] };
    v2f b2 = { w0[32], w1[32] };
    v2f b3 = { w0[48], w1[48] };

    // 8 args: (neg_a, A, neg_b, B, c_mod, C, reuse_a, reuse_b)
    c0 = __builtin_amdgcn_wmma_f32_16x16x4_f32(false, a, false, b0, (short)0, c0, false, false);
    c1 = __builtin_amdgcn_wmma_f32_16x16x4_f32(false, a, false, b1, (short)0, c1, false, false);
    c2 = __builtin_amdgcn_wmma_f32_16x16x4_f32(false, a, false, b2, (short)0, c2, false, false);
    c3 = __builtin_amdgcn_wmma_f32_16x16x4_f32(false, a, false, b3, (short)0, c3, false, false);
  }

  float* __restrict__ out = H + (size_t)row0 * NC;
#pragma unroll
  for (int r = 0; r < 8; ++r) {
    size_t m = (size_t)(r + half * 8) * NC;
    out[m +  0 + l15] = c0[r];
    out[m + 16 + l15] = c1[r];
    out[m + 32 + l15] = c2[r];
    out[m + 48 + l15] = c3[r];
  }
}

// ---------------------------------------------------------------------------
// Degree / normalization helpers
// ---------------------------------------------------------------------------
__global__ __launch_bounds__(256) void init_deg(float* __restrict__ deg, int n) {
  int i = blockIdx.x * 256 + threadIdx.x;
  if (i < n) deg[i] = 1.0f;  // self-loop contribution
}

__global__ __launch_bounds__(256) void accum_deg(const int* __restrict__ dst,
                                                 float* __restrict__ deg, int e) {
  int i = blockIdx.x * 256 + threadIdx.x;
  if (i < e) atomicAdd(&deg[dst[i]], 1.0f);
}

__global__ __launch_bounds__(256) void finalize_dinv(float* __restrict__ deg, int n) {
  int i = blockIdx.x * 256 + threadIdx.x;
  if (i < n) {
    float d = deg[i];
    deg[i] = (d > 0.0f) ? rsqrtf(d) : 0.0f;
  }
}

__global__ __launch_bounds__(256) void compute_norm(const int* __restrict__ src,
                                                    const int* __restrict__ dst,
                                                    const float* __restrict__ dinv,
                                                    float* __restrict__ norm, int e) {
  int i = blockIdx.x * 256 + threadIdx.x;
  if (i < e) norm[i] = dinv[src[i]] * dinv[dst[i]];
}

// ---------------------------------------------------------------------------
// Aggregation init: out[i][c] = bias[c] + h[i][c]*dinv[i]^2  (self-loop + bias)
// float4-vectorized: 16 lanes per node row, B128 loads/stores.
// ---------------------------------------------------------------------------
__global__ __launch_bounds__(256) void agg_init(const float* __restrict__ h,
                                                const float* __restrict__ dinv,
                                                const float* __restrict__ bias,
                                                float* __restrict__ out, int n) {
  size_t tid = (size_t)blockIdx.x * 256 + threadIdx.x;
  int i = (int)(tid >> 4);          // node
  int q = (int)(tid & 15);          // float4 slot within the 64-wide row
  if (i < n) {
    float di = dinv[i];
    float s = di * di;
    float4 hv = ((const float4*)(h + (size_t)i * NC))[q];
    float4 bv = ((const float4*)bias)[q];
    float4 r;
    r.x = bv.x + hv.x * s;
    r.y = bv.y + hv.y * s;
    r.z = bv.z + hv.z * s;
    r.w = bv.w + hv.w * s;
    ((float4*)(out + (size_t)i * NC))[q] = r;
  }
}

// ---------------------------------------------------------------------------
// Edge scatter: out[dst[e]][c] += h[src[e]][c] * norm[e]
// 32 consecutive lanes per edge, float2 (B64) gathers -> coalesced 256B rows.
// Per-layer working set (~29MB) fits in the 192MB L2, so the fp32 atomics
// resolve on-chip.
// ---------------------------------------------------------------------------
__global__ __launch_bounds__(256) void edge_agg(const float* __restrict__ h,
                                                const int* __restrict__ src,
                                                const int* __restrict__ dst,
                                                const float* __restrict__ norm,
                                                float* __restrict__ out, int e) {
  size_t tid = (size_t)blockIdx.x * 256 + threadIdx.x;
  int ed = (int)(tid >> 5);         // edge
  int c2 = (int)(tid & 31);         // float2 slot within the 64-wide row
  if (ed < e) {
    int s = src[ed];
    int d = dst[ed];
    float nv = norm[ed];
    float2 v = ((const float2*)(h + (size_t)s * NC))[c2];
    float* op = out + (size_t)d * NC + (size_t)c2 * 2;
    atomicAdd(op,     v.x * nv);
    atomicAdd(op + 1, v.y * nv);
  }
}

// ---------------------------------------------------------------------------
extern "C" void kernel_launch(void* const* d_in, const int* in_sizes, int n_in,
                              void* d_out, int out_size, void* d_ws, size_t ws_size,
                              hipStream_t stream) {
  const float* x  = (const float*)d_in[0];
  const int*   ei = (const int*)d_in[1];
  const float* W1 = (const float*)d_in[2];
  const float* b1 = (const float*)d_in[3];
  const float* W2 = (const float*)d_in[4];
  const float* b2 = (const float*)d_in[5];
  const float* W3 = (const float*)d_in[6];
  const float* b3 = (const float*)d_in[7];
  float* out = (float*)d_out;

  const int IN_DIM = 128;
  const int N = in_sizes[0] / IN_DIM;   // 50000
  const int E = in_sizes[1] / 2;        // 800000
  const int* src = ei;
  const int* dst = ei + E;

  // workspace carve-out (256B aligned)
  char* ws = (char*)d_ws;
  size_t off = 0;
  auto carve = [&](size_t bytes) -> float* {
    float* p = (float*)(ws + off);
    off = (off + bytes + 255) & ~(size_t)255;
    return p;
  };
  float* dinv = carve((size_t)N * sizeof(float));        // 0.2 MB
  float* norm = carve((size_t)E * sizeof(float));        // 3.2 MB
  float* Hbuf = carve((size_t)N * NC * sizeof(float));   // 12.8 MB (post-GEMM)
  float* Abuf = carve((size_t)N * NC * sizeof(float));   // 12.8 MB (aggregated)

  const int nodeBlocks  = (N + 255) / 256;
  const int edgeBlocks  = (E + 255) / 256;
  const int nodeVBlocks = (int)(((size_t)N * 16 + 255) / 256);  // float4 lanes
  const int edgeVBlocks = (int)(((size_t)E * 32 + 255) / 256);  // float2 lanes
  const int gemmBlocks  = (N + 15) / 16;

  // --- normalization: deg -> dinv -> per-edge norm ---
  init_deg<<<nodeBlocks, 256, 0, stream>>>(dinv, N);
  accum_deg<<<edgeBlocks, 256, 0, stream>>>(dst, dinv, E);
  finalize_dinv<<<nodeBlocks, 256, 0, stream>>>(dinv, N);
  compute_norm<<<edgeBlocks, 256, 0, stream>>>(src, dst, dinv, norm, E);

  // --- layer 1: H = X @ W1 ; A = agg(H) + b1 ---
  gcn_gemm_wmma_f32<0><<<gemmBlocks, 32, 0, stream>>>(x, W1, Hbuf, N, IN_DIM);
  agg_init<<<nodeVBlocks, 256, 0, stream>>>(Hbuf, dinv, b1, Abuf, N);
  edge_agg<<<edgeVBlocks, 256, 0, stream>>>(Hbuf, src, dst, norm, Abuf, E);

  // --- layer 2: H = relu(A) @ W2 ; A = agg(H) + b2 ---
  gcn_gemm_wmma_f32<1><<<gemmBlocks, 32, 0, stream>>>(Abuf, W2, Hbuf, N, NC);
  agg_init<<<nodeVBlocks, 256, 0, stream>>>(Hbuf, dinv, b2, Abuf, N);
  edge_agg<<<edgeVBlocks, 256, 0, stream>>>(Hbuf, src, dst, norm, Abuf, E);

  // --- layer 3: H = relu(A) @ W3 ; out = agg(H) + b3 ---
  gcn_gemm_wmma_f32<1><<<gemmBlocks, 32, 0, stream>>>(Abuf, W3, Hbuf, N, NC);
  agg_init<<<nodeVBlocks, 256, 0, stream>>>(Hbuf, dinv, b3, out, N);
  edge_agg<<<edgeVBlocks, 256, 0, stream>>>(Hbuf, src, dst, norm, out, E);
}